// MultiBoxLoss_84593675862034
// MI455X (gfx1250) — compile-verified
//
#include <hip/hip_runtime.h>
#include <math.h>

#define BATCH 64
#define NANCH 8732
#define NDIM  85
#define NCLS  81
#define NGT   32
#define PPAD  8960              // NANCH padded up to a multiple of 128 (70*128)

typedef __attribute__((ext_vector_type(2))) float        v2f;
typedef __attribute__((ext_vector_type(8))) float        v8f;
typedef __attribute__((ext_vector_type(4))) unsigned int v4u;
typedef __attribute__((ext_vector_type(8))) int          v8i;
typedef __attribute__((ext_vector_type(4))) int          v4i;

__device__ __forceinline__ float wave_max32(float v) {
  #pragma unroll
  for (int off = 16; off; off >>= 1) v = fmaxf(v, __shfl_xor(v, off, 32));
  return v;
}
__device__ __forceinline__ float wave_sum32(float v) {
  #pragma unroll
  for (int off = 16; off; off >>= 1) v += __shfl_xor(v, off, 32);
  return v;
}

// ---------------------------------------------------------------- K0: zero
__global__ void mb_zero_kernel(float* __restrict__ out, int* __restrict__ npos,
                               float* __restrict__ pospart, int ppn) {
  const int i = blockIdx.x * blockDim.x + threadIdx.x;
  if (i == 0) out[0] = 0.0f;
  if (i < BATCH) npos[i] = 0;
  for (int j = i; j < ppn; j += gridDim.x * blockDim.x) pospart[j] = 0.0f;
}

// ---------------------------------------------------------------- K1: match
__global__ void mb_match_kernel(const float* __restrict__ anchors,
                                const float* __restrict__ gt_boxes,
                                const int*   __restrict__ gt_labels,
                                const int*   __restrict__ gt_counts,
                                float* __restrict__ tloc,
                                float* __restrict__ tcls) {
  const int b   = blockIdx.x;
  const int tid = threadIdx.x;

  __shared__ float gx1[NGT], gy1[NGT], gx2[NGT], gy2[NGT], garea[NGT];
  __shared__ float gcx[NGT], gcy[NGT], gww[NGT], ghh[NGT];
  __shared__ int   glab[NGT], gval[NGT];
  __shared__ unsigned long long gbest[NGT];

  const int count = gt_counts[b];
  if (tid < NGT) {
    const float* g = gt_boxes + ((size_t)b * NGT + tid) * 4;
    float cx = g[0], cy = g[1], w = g[2], h = g[3];
    gcx[tid] = cx; gcy[tid] = cy; gww[tid] = w; ghh[tid] = h;
    float x1 = cx - 0.5f * w, y1 = cy - 0.5f * h;
    float x2 = cx + 0.5f * w, y2 = cy + 0.5f * h;
    gx1[tid] = x1; gy1[tid] = y1; gx2[tid] = x2; gy2[tid] = y2;
    garea[tid] = (x2 - x1) * (y2 - y1);
    glab[tid] = gt_labels[b * NGT + tid];
    gval[tid] = (tid < count) ? 1 : 0;
    gbest[tid] = 0ull;
  }
  __syncthreads();

  // per-thread running best anchor per GT (register arrays, fully unrolled)
  float    lbv[NGT];
  unsigned lba[NGT];
  #pragma unroll
  for (int g = 0; g < NGT; ++g) { lbv[g] = -1.0f; lba[g] = 0u; }

  for (int a = tid; a < NANCH; a += blockDim.x) {
    float4 an = ((const float4*)anchors)[a];
    float acx = an.x, acy = an.y, aw = an.z, ah = an.w;
    float ax1 = acx - 0.5f * aw, ay1 = acy - 0.5f * ah;
    float ax2 = acx + 0.5f * aw, ay2 = acy + 0.5f * ah;
    float aarea = (ax2 - ax1) * (ay2 - ay1);

    float bestv = -1e30f; int bestg = 0;
    #pragma unroll
    for (int g = 0; g < NGT; ++g) {
      float lx = fmaxf(ax1, gx1[g]), ly = fmaxf(ay1, gy1[g]);
      float rx = fminf(ax2, gx2[g]), ry = fminf(ay2, gy2[g]);
      float iw = fmaxf(rx - lx, 0.0f), ih = fmaxf(ry - ly, 0.0f);
      float inter = iw * ih;
      float iou = inter / (aarea + garea[g] - inter);
      iou = gval[g] ? iou : -1.0f;
      if (iou > bestv) { bestv = iou; bestg = g; }          // first-max
      if (iou > lbv[g]) { lbv[g] = iou; lba[g] = (unsigned)a; }
    }

    bool  over = bestv > 0.5f;
    float l0 = 0.f, l1 = 0.f, l2 = 0.f, l3 = 0.f, cls = 0.f;
    if (over) {
      cls = (float)(glab[bestg] + 1);
      l0 = (gcx[bestg] - acx) / aw;
      l1 = (gcy[bestg] - acy) / ah;
      l2 = logf(gww[bestg]) - logf(aw);
      l3 = logf(ghh[bestg]) - logf(ah);
    }
    size_t o = (size_t)b * NANCH + a;
    ((float4*)tloc)[o] = make_float4(l0, l1, l2, l3);
    tcls[o] = cls;
  }

  // fold per-thread bests: key = iou_bits<<32 | ~a  (max => best iou, smallest a)
  #pragma unroll
  for (int g = 0; g < NGT; ++g) {
    unsigned long long key =
        ((unsigned long long)__float_as_uint(fmaxf(lbv[g], 0.0f)) << 32) |
        (unsigned long long)(0xFFFFFFFFu - lba[g]);
    atomicMax(&gbest[g], key);
  }
  __syncthreads();

  // forced matches: every valid GT claims its best anchor
  if (tid < NGT && tid < count) {
    unsigned abest = 0xFFFFFFFFu - (unsigned)(gbest[tid] & 0xFFFFFFFFull);
    float4 an = ((const float4*)anchors)[abest];
    size_t o = (size_t)b * NANCH + abest;
    float l0 = (gcx[tid] - an.x) / an.z;
    float l1 = (gcy[tid] - an.y) / an.w;
    float l2 = logf(gww[tid]) - logf(an.z);
    float l3 = logf(ghh[tid]) - logf(an.w);
    ((float4*)tloc)[o] = make_float4(l0, l1, l2, l3);
    tcls[o] = (float)(glab[tid] + 1);
  }
}

// ---------------------------------------------------------------- K2: CE (wave per anchor)
__global__ void mb_ce_kernel(const float* __restrict__ pred,
                             const float* __restrict__ tloc,
                             const float* __restrict__ tcls,
                             float* __restrict__ pospart,
                             float* __restrict__ allneg,
                             int*   __restrict__ npos) {
  const int lane = threadIdx.x & 31;
  const int wid  = blockIdx.x * (blockDim.x >> 5) + (threadIdx.x >> 5);
  if (wid >= BATCH * NANCH) return;
  const int b = wid / NANCH;
  const int a = wid - b * NANCH;

  const float* p = pred + (size_t)wid * NDIM;
  __builtin_prefetch(p + NDIM * (blockDim.x >> 5), 0, 0);  // global_prefetch_b8

  // 81 classes across 32 lanes: 2 full strides + 17-lane tail (branchless load)
  float c0 = p[4 + lane];
  float c1 = p[36 + lane];                         // lane+32 < 81 always
  const int  i2  = (lane < 17) ? lane : 16;        // clamped, always in-bounds
  const bool m2  = (lane < 17);
  float c2v = p[68 + i2];
  float c2  = m2 ? c2v : -__builtin_inff();

  float m = wave_max32(fmaxf(c0, fmaxf(c1, c2)));
  float e = expf(c0 - m) + expf(c1 - m) + (m2 ? expf(c2 - m) : 0.0f);
  float logZ = logf(wave_sum32(e));

  int tc   = (int)tcls[wid];          // 0..80, uniform across wave
  int slot = tc >> 5, src = tc & 31;
  float sel = (slot == 0) ? c0 : ((slot == 1) ? c1 : c2);
  float xt  = __shfl(sel, src, 32);
  float ce  = (m + logZ) - xt;        // -log_softmax[tc]
  bool  pos = tc > 0;

  float d = 0.0f;
  if (lane < 4) {
    float dd = p[lane] - tloc[(size_t)wid * 4 + lane];
    float ad = fabsf(dd);
    d = (ad < 1.0f) ? 0.5f * dd * dd : ad - 0.5f;
  }
  float locsum = wave_sum32(d);

  if (lane == 0) {
    pospart[(size_t)b * PPAD + a] = pos ? (locsum + ce) : 0.0f;
    allneg[wid]                   = pos ? 0.0f : ce;
    if (pos) atomicAdd(&npos[b], 1);
  }
}

// ---------------------------------------------------------------- K3: mining + WMMA reduce
__global__ void mb_reduce_kernel(const float* __restrict__ allneg,
                                 const float* __restrict__ pospart,
                                 const int*   __restrict__ npos,
                                 float* __restrict__ out) {
  extern __shared__ float s[];        // 16384 floats (64 KB dynamic LDS)
  __shared__ float wsum[32];
  const int b = blockIdx.x, tid = threadIdx.x;
  const int N = 16384;

#if __has_builtin(__builtin_amdgcn_tensor_load_to_lds)
  // TDM: DMA all_neg[b][0..NANCH) (34.9 KB) global -> LDS, wave 0 issues.
  if ((tid >> 5) == 0) {              // whole wave 0 active (EXEC all-ones)
    unsigned lds_off = (unsigned)(size_t)(void*)s;
    unsigned long long ga =
        (unsigned long long)(size_t)(const void*)(allneg + (size_t)b * NANCH);
    v4u g0;
    g0[0] = 1u;                                       // count=1 (valid), user mode
    g0[1] = lds_off;                                  // lds_addr
    g0[2] = (unsigned)(ga & 0xFFFFFFFFull);           // global_addr[31:0]
    g0[3] = (unsigned)((ga >> 32) & 0x01FFFFFFull)    // global_addr[56:32]
            | 0x80000000u;                            // type=2 ("image")
    v8i g1;
    g1[0] = 0x00020000;                               // data_size=4B, no mcast
    g1[1] = (int)(((unsigned)NANCH & 0xFFFFu) << 16); // tensor_dim0[15:0]
    g1[2] = (int)((((unsigned)NANCH >> 16) & 0xFFFFu) // tensor_dim0[31:16]
            | (1u << 16));                            // tensor_dim1=1
    g1[3] = (int)(((unsigned)NANCH & 0xFFFFu) << 16); // tile_dim0=NANCH
    g1[4] = 1;                                        // tile_dim1=1, tile_dim2=0
    g1[5] = NANCH;                                    // tensor_dim0_stride[31:0]
    g1[6] = (int)(((unsigned)NANCH & 0xFFFFu) << 16); // dim1_stride[15:0]
    g1[7] = 0;
    v4i gz  = {0, 0, 0, 0};
    v8i gz8 = {0, 0, 0, 0, 0, 0, 0, 0};
    __builtin_amdgcn_tensor_load_to_lds(g0, g1, gz, gz, gz8, 0);
    __builtin_amdgcn_s_wait_tensorcnt(0);
  }
  for (int i = NANCH + tid; i < N; i += blockDim.x) s[i] = -1.0f;  // pad tail
  __syncthreads();
#else
  for (int i = tid; i < N; i += blockDim.x)
    s[i] = (i < NANCH) ? allneg[(size_t)b * NANCH + i] : -1.0f;
  __syncthreads();
#endif

  // bitonic sort, descending
  for (int k = 2; k <= N; k <<= 1) {
    for (int j = k >> 1; j > 0; j >>= 1) {
      for (int i = tid; i < N; i += blockDim.x) {
        int partner = i ^ j;
        if (partner > i) {
          float x = s[i], y = s[partner];
          bool desc = ((i & k) == 0);
          if (desc ? (x < y) : (x > y)) { s[i] = y; s[partner] = x; }
        }
      }
      __syncthreads();
    }
  }

  const int np      = npos[b];
  const int num_neg = min(3 * np, NANCH - 1);
  float negp = 0.0f;
  for (int i = tid; i < num_neg; i += blockDim.x) negp += s[i];
  __syncthreads();
  s[tid] = negp;
  __syncthreads();
  for (int st = blockDim.x >> 1; st > 0; st >>= 1) {
    if (tid < st) s[tid] += s[tid + st];
    __syncthreads();
  }

  // positive-part sum via V_WMMA_F32_16X16X4_F32: D = ones(16x4) x B + C.
  // pospart is padded+zeroed to PPAD per batch -> branchless b64 loads,
  // two accumulators keep 2 loads in flight per WMMA pair.
  const int lane = tid & 31, w = tid >> 5, nw = blockDim.x >> 5;
  v2f ones; ones[0] = 1.0f; ones[1] = 1.0f;
  v8f acc0 = {}, acc1 = {};
  const float* pp = pospart + (size_t)b * PPAD;
  for (int base = w * 128; base < PPAD; base += nw * 128) {
    float2 t0 = ((const float2*)(pp + base))[lane];
    float2 t1 = ((const float2*)(pp + base + 64))[lane];
    v2f b0; b0[0] = t0.x; b0[1] = t0.y;
    v2f b1; b1[0] = t1.x; b1[1] = t1.y;
    acc0 = __builtin_amdgcn_wmma_f32_16x16x4_f32(false, ones, false, b0,
                                                 (short)0, acc0, false, false);
    acc1 = __builtin_amdgcn_wmma_f32_16x16x4_f32(false, ones, false, b1,
                                                 (short)0, acc1, false, false);
  }
  v8f acc = acc0 + acc1;
  float sb = acc[0] + acc[1] + acc[2] + acc[3] + acc[4] + acc[5] + acc[6] + acc[7];
  sb = wave_sum32(sb);                       // = 16 * wave_total
  if (lane == 0) wsum[w] = sb * (1.0f / 16.0f);
  __syncthreads();

  if (tid == 0) {
    float possum = 0.0f;
    for (int i = 0; i < nw; ++i) possum += wsum[i];
    float loss = (s[0] + possum) / (float)np;
    atomicAdd(out, loss);
  }
}

// ---------------------------------------------------------------- launcher
extern "C" void kernel_launch(void* const* d_in, const int* in_sizes, int n_in,
                              void* d_out, int out_size, void* d_ws, size_t ws_size,
                              hipStream_t stream) {
  (void)in_sizes; (void)n_in; (void)out_size; (void)ws_size;
  const float* pred      = (const float*)d_in[0];
  const float* anchors   = (const float*)d_in[1];
  const float* gt_boxes  = (const float*)d_in[2];
  const int*   gt_labels = (const int*)d_in[3];
  const int*   gt_counts = (const int*)d_in[4];
  float* out = (float*)d_out;

  const size_t BA = (size_t)BATCH * NANCH;
  float* ws      = (float*)d_ws;
  float* tloc    = ws;                    // 4*BA floats
  float* tcls    = tloc + 4 * BA;         // BA floats
  float* pospart = tcls + BA;             // BATCH*PPAD floats (padded, zeroed)
  float* allneg  = pospart + (size_t)BATCH * PPAD;  // BA floats
  int*   npos    = (int*)(allneg + BA);   // BATCH ints

  mb_zero_kernel<<<1024, 256, 0, stream>>>(out, npos, pospart, BATCH * PPAD);
  mb_match_kernel<<<BATCH, 256, 0, stream>>>(anchors, gt_boxes, gt_labels,
                                             gt_counts, tloc, tcls);
  const int totalw = BATCH * NANCH;
  const int blocks = (totalw + 7) / 8;    // 8 waves (256 thr) per block
  mb_ce_kernel<<<blocks, 256, 0, stream>>>(pred, tloc, tcls, pospart, allneg, npos);
  mb_reduce_kernel<<<BATCH, 1024, 16384 * sizeof(float), stream>>>(allneg, pospart,
                                                                   npos, out);
}